// BatchCRFLoss_77343771067068
// MI455X (gfx1250) — compile-verified
//
#include <hip/hip_runtime.h>
#include <hip/hip_bf16.h>

// Problem constants (from reference)
#define NB   128      // batch
#define CC   16       // states
#define LL   16384    // sequence length
#define SEG  64       // number of scan segments per batch
#define SLEN (LL/SEG) // 256 steps per segment

typedef float v8f __attribute__((ext_vector_type(8)));
typedef float v2f __attribute__((ext_vector_type(2)));
typedef unsigned int uint2v __attribute__((ext_vector_type(2)));

// ---------- v_permlane16_swap_b32: swap odd 16-lane row of a with even row of b
// returns (a' = [a.lo | b.lo], b' = [a.hi | b.hi]) ----------
__device__ __forceinline__ void plswap(float& a, float& b) {
    uint2v r = __builtin_amdgcn_permlane16_swap(__float_as_uint(a), __float_as_uint(b),
                                                false, false);
    a = __uint_as_float(r.x);
    b = __uint_as_float(r.y);
}

// cross-half column sum: every lane gets (v[lane] + v[lane^16])
__device__ __forceinline__ float halfsum(float v) {
    float x = v, y = v;
    plswap(x, y);   // x = [lo|lo], y = [hi|hi]
    return x + y;
}

// =====================================================================
// Kernel 1: eobsT[n][t][c] = exp(observes[n][c][t])   (coalesced transpose)
// =====================================================================
__global__ __launch_bounds__(256) void make_eobsT(const float* __restrict__ obs,
                                                  float* __restrict__ eobsT) {
    __shared__ float tile[16][64 + 1];
    const int n  = blockIdx.y;
    const int t0 = blockIdx.x * 64;
    const int tid = threadIdx.x;
    for (int e = tid; e < 16 * 64; e += 256) {
        int c = e / 64, tt = e % 64;
        tile[c][tt] = obs[((size_t)n * 16 + c) * LL + t0 + tt];
    }
    __syncthreads();
    for (int e = tid; e < 16 * 64; e += 256) {
        int tt = e / 16, c = e % 16;
        eobsT[(((size_t)n * LL) + t0 + tt) * 16 + c] = __expf(tile[c][tt]);
    }
}

// =====================================================================
// Kernel 2: segmented forward scan. One wave32 per (n, segment).
// State P (prob-space 16x16 matrix) in WMMA C/D layout:
//   reg r, lanes 0-15 -> row r, col=lane; lanes 16-31 -> row r+8.
// Step: P <- diag(eobs_t) * E * P, E = exp(T) (A-layout, preloaded).
// Per-column renorm every 8 steps with carried log-offset bcol.
// Output: segMat[n][s][j][i] = log(P[j,i]) + bcol[i]
// =====================================================================
__device__ __forceinline__ void loadblk(const float4* __restrict__ ep, int t,
                                        float4* dst) {
#pragma unroll
    for (int u = 0; u < 8; ++u) {
        dst[2 * u]     = ep[(size_t)(t + u) * 4];
        dst[2 * u + 1] = ep[(size_t)(t + u) * 4 + 1];
    }
}

__global__ __launch_bounds__(32) void seg_scan(const float* __restrict__ eobsT,
                                               const float* __restrict__ trans,
                                               const int*   __restrict__ lengths,
                                               float* __restrict__ segMat) {
    const int tid = threadIdx.x;
    const int wid = blockIdx.x;          // 0 .. NB*SEG-1
    const int n   = wid / SEG;
    const int s   = wid % SEG;
    const int h   = (tid >> 4) & 1;      // lane half
    const int col = tid & 15;

    // E in WMMA A-layout: chunk c, elem q, half h holds E[row, 4c+q+2h], row = lane&15
    v2f Ea[4];
#pragma unroll
    for (int c = 0; c < 4; ++c) {
        Ea[c][0] = __expf(trans[col * 16 + (4 * c + 0 + 2 * h)]);
        Ea[c][1] = __expf(trans[col * 16 + (4 * c + 1 + 2 * h)]);
    }

    // state = identity
    float m[8];
#pragma unroll
    for (int r = 0; r < 8; ++r) m[r] = ((r + 8 * h) == col) ? 1.0f : 0.0f;
    float bcol = 0.0f;

    const int len  = lengths[n];
    const int t0   = s * SLEN;
    const int tend = min(t0 + SLEN, len);   // steps with t>=len are identity -> skip

    // one scan step: P <- diag(e) * E * P
    auto step = [&](float4 e0, float4 e1) {
        // relayout D-layout state -> WMMA B-layout chunks via permlane16_swap
        float a0 = m[0], a2 = m[2]; plswap(a0, a2);   // a0=[r0|r2],  a2=[r8|r10]
        float a1 = m[1], a3 = m[3]; plswap(a1, a3);   // a1=[r1|r3],  a3=[r9|r11]
        float a4 = m[4], a6 = m[6]; plswap(a4, a6);   // a4=[r4|r6],  a6=[r12|r14]
        float a5 = m[5], a7 = m[7]; plswap(a5, a7);   // a5=[r5|r7],  a7=[r13|r15]
        v2f B0 = {a0, a1}, B1 = {a4, a5}, B2 = {a2, a3}, B3 = {a6, a7};
        v8f acc = {};
        acc = __builtin_amdgcn_wmma_f32_16x16x4_f32(false, Ea[0], false, B0, (short)0, acc, false, false);
        acc = __builtin_amdgcn_wmma_f32_16x16x4_f32(false, Ea[1], false, B1, (short)0, acc, false, false);
        acc = __builtin_amdgcn_wmma_f32_16x16x4_f32(false, Ea[2], false, B2, (short)0, acc, false, false);
        acc = __builtin_amdgcn_wmma_f32_16x16x4_f32(false, Ea[3], false, B3, (short)0, acc, false, false);
        m[0] = acc[0] * e0.x; m[1] = acc[1] * e0.y; m[2] = acc[2] * e0.z; m[3] = acc[3] * e0.w;
        m[4] = acc[4] * e1.x; m[5] = acc[5] * e1.y; m[6] = acc[6] * e1.z; m[7] = acc[7] * e1.w;
    };

    // per-lane eobs stream: 8 floats (rows 8h..8h+7) at (n*LL+t)*16 + 8h
    const float4* ep = (const float4*)eobsT + ((size_t)n * LL) * 4 + 2 * h;

    int t = t0;
    const int tfull = t0 + (((tend - t0) >> 3) << 3);   // whole 8-step blocks
    float4 eb[16];
    if (t < tfull) loadblk(ep, t, eb);
    for (; t < tfull; t += 8) {
        float4 nb[16];
        const bool more = (t + 8) < tfull;
        if (more) loadblk(ep, t + 8, nb);               // double-buffer next block
#pragma unroll
        for (int u = 0; u < 8; ++u) step(eb[2 * u], eb[2 * u + 1]);
        // per-column renormalization (keeps f32 in range); fv = bcol + log p invariant
        float sp = ((m[0] + m[1]) + (m[2] + m[3])) + ((m[4] + m[5]) + (m[6] + m[7]));
        float ss = halfsum(sp);
        float inv = __builtin_amdgcn_rcpf(ss);
#pragma unroll
        for (int r = 0; r < 8; ++r) m[r] *= inv;
        bcol += __logf(ss);
        if (more) {
#pragma unroll
            for (int i = 0; i < 16; ++i) eb[i] = nb[i];
        }
    }
    for (; t < tend; ++t) {                              // <=7 tail steps, no renorm needed
        float4 e0 = ep[(size_t)t * 4];
        float4 e1 = ep[(size_t)t * 4 + 1];
        step(e0, e1);
    }

    // emit log-semiring segment matrix
    float* outp = segMat + (size_t)(n * SEG + s) * 256;
#pragma unroll
    for (int r = 0; r < 8; ++r) {
        float v = fmaxf(__logf(m[r]), -1e30f) + bcol;   // clamp log(0) -> -1e30
        outp[(r + 8 * h) * 16 + col] = v;
    }
}

// =====================================================================
// Kernel 3: per-batch tree combine of SEG log-matrices, then forward_score
//   C[j,i] = logsumexp_k( A[j,k] + B[k,i] )   (A = later segment)
// =====================================================================
__device__ __forceinline__ float lse16(const float* __restrict__ Ar,
                                       const float* __restrict__ Bc) {
    float mx = Ar[0] + Bc[0];
#pragma unroll
    for (int k = 1; k < 16; ++k) mx = fmaxf(mx, Ar[k] + Bc[k * 16]);
    float sum = 0.0f;
#pragma unroll
    for (int k = 0; k < 16; ++k) sum += __expf(Ar[k] + Bc[k * 16] - mx);
    return mx + __logf(sum);
}

__device__ __forceinline__ void combine_level(const float* __restrict__ src,
                                              float* __restrict__ dst,
                                              int outCount, int tid) {
    for (int e = tid; e < outCount * 256; e += 256) {
        int mm = e >> 8, j = (e >> 4) & 15, i = e & 15;
        const float* A = src + (2 * mm + 1) * 256;
        const float* B = src + (2 * mm) * 256;
        dst[e] = lse16(A + j * 16, B + i);
    }
}

__global__ __launch_bounds__(256) void combine_kernel(const float* __restrict__ segMat,
                                                      float* __restrict__ fs) {
    __shared__ float bufA[32 * 256];   // 32 KB
    __shared__ float bufB[16 * 256];   // 16 KB
    __shared__ float red[256];
    const int n = blockIdx.x;
    const int tid = threadIdx.x;
    const float* base = segMat + (size_t)n * SEG * 256;

    // level 1: 64 -> 32 (global -> LDS)
    for (int e = tid; e < 32 * 256; e += 256) {
        int mm = e >> 8, j = (e >> 4) & 15, i = e & 15;
        const float* A = base + (size_t)(2 * mm + 1) * 256;
        const float* B = base + (size_t)(2 * mm) * 256;
        bufA[e] = lse16(A + j * 16, B + i);
    }
    __syncthreads();
    combine_level(bufA, bufB, 16, tid); __syncthreads();
    combine_level(bufB, bufA,  8, tid); __syncthreads();
    combine_level(bufA, bufB,  4, tid); __syncthreads();
    combine_level(bufB, bufA,  2, tid); __syncthreads();
    combine_level(bufA, bufB,  1, tid); __syncthreads();

    // forward_score[n] = LSE over all 256 entries (fv0 = 0)
    float v = bufB[tid];
    red[tid] = v; __syncthreads();
    for (int st = 128; st > 0; st >>= 1) {
        if (tid < st) red[tid] = fmaxf(red[tid], red[tid + st]);
        __syncthreads();
    }
    float mx = red[0]; __syncthreads();
    red[tid] = __expf(v - mx); __syncthreads();
    for (int st = 128; st > 0; st >>= 1) {
        if (tid < st) red[tid] += red[tid + st];
        __syncthreads();
    }
    if (tid == 0) fs[n] = mx + __logf(red[0]);
}

// =====================================================================
// Kernel 4: tags[n][t] = argmax_c answers[n][c][t]  (first max, like jnp.argmax)
// =====================================================================
__global__ __launch_bounds__(256) void tags_kernel(const float* __restrict__ ans,
                                                   int* __restrict__ tags) {
    const int n = blockIdx.y;
    const int t = blockIdx.x * 256 + threadIdx.x;
    const float* a = ans + (size_t)n * 16 * LL + t;
    float best = a[0];
    int bi = 0;
#pragma unroll
    for (int c = 1; c < 16; ++c) {
        float v = a[(size_t)c * LL];
        if (v > best) { best = v; bi = c; }
    }
    tags[(size_t)n * LL + t] = bi;
}

// =====================================================================
// Kernel 5: gold[n] = sum_{t<min(L-1,len)} T[tag[t],tag[t+1]] + obs[n,tag[t+1],t]
// =====================================================================
__global__ __launch_bounds__(256) void gold_kernel(const float* __restrict__ obs,
                                                   const float* __restrict__ trans,
                                                   const int*   __restrict__ tags,
                                                   const int*   __restrict__ lengths,
                                                   float* __restrict__ gold) {
    __shared__ float red[256];
    const int n = blockIdx.x;
    const int tid = threadIdx.x;
    const int tmax = min(LL - 1, lengths[n]);
    const int* tg = tags + (size_t)n * LL;
    const float* ob = obs + (size_t)n * 16 * LL;
    float acc = 0.0f;
    for (int t = tid; t < tmax; t += 256) {
        int cur = tg[t], nxt = tg[t + 1];
        acc += trans[cur * 16 + nxt] + ob[(size_t)nxt * LL + t];
    }
    red[tid] = acc; __syncthreads();
    for (int st = 128; st > 0; st >>= 1) {
        if (tid < st) red[tid] += red[tid + st];
        __syncthreads();
    }
    if (tid == 0) gold[n] = red[0];
}

// =====================================================================
// Kernel 6: out = sum_n (fs[n] - gold[n]) / N
// =====================================================================
__global__ __launch_bounds__(128) void final_kernel(const float* __restrict__ fs,
                                                    const float* __restrict__ gold,
                                                    float* __restrict__ out) {
    __shared__ float red[128];
    const int tid = threadIdx.x;
    red[tid] = fs[tid] - gold[tid];
    __syncthreads();
    for (int st = 64; st > 0; st >>= 1) {
        if (tid < st) red[tid] += red[tid + st];
        __syncthreads();
    }
    if (tid == 0) out[0] = red[0] / (float)NB;
}

// =====================================================================
extern "C" void kernel_launch(void* const* d_in, const int* in_sizes, int n_in,
                              void* d_out, int out_size, void* d_ws, size_t ws_size,
                              hipStream_t stream) {
    const float* obs     = (const float*)d_in[0];   // (N,C,L) f32
    const float* ans     = (const float*)d_in[1];   // (N,C,L) f32
    const float* trans   = (const float*)d_in[2];   // (C,C)   f32
    const int*   lengths = (const int*)  d_in[3];   // (N,)    i32
    float* out = (float*)d_out;

    // workspace layout
    float* eobsT  = (float*)d_ws;                               // N*L*16 floats (128 MB)
    float* segMat = eobsT + (size_t)NB * LL * 16;               // N*SEG*256 floats (8 MB)
    int*   tags   = (int*)(segMat + (size_t)NB * SEG * 256);    // N*L ints (8 MB)
    float* fs     = (float*)(tags + (size_t)NB * LL);           // N floats
    float* gold   = fs + NB;                                    // N floats

    make_eobsT    <<<dim3(LL / 64, NB), 256, 0, stream>>>(obs, eobsT);
    seg_scan      <<<NB * SEG, 32, 0, stream>>>(eobsT, trans, lengths, segMat);
    combine_kernel<<<NB, 256, 0, stream>>>(segMat, fs);
    tags_kernel   <<<dim3(LL / 256, NB), 256, 0, stream>>>(ans, tags);
    gold_kernel   <<<NB, 256, 0, stream>>>(obs, trans, tags, lengths, gold);
    final_kernel  <<<1, 128, 0, stream>>>(fs, gold, out);
}